// GNNLayer_89146341196447
// MI455X (gfx1250) — compile-verified
//
#include <hip/hip_runtime.h>
#include <stdint.h>

typedef __attribute__((ext_vector_type(2))) float v2f;
typedef __attribute__((ext_vector_type(8))) float v8f;

#define D_FEAT 128
#define K_FUSED 256
#define LDS_STRIDE 260   // row*260+k: bank = (4*row + k) % 64 -> conflict-free A-frag loads
#define NEG_SLOPE 0.01f

// ---------------------------------------------------------------------------
// Kernel 0: zero h_neigh accumulator (float4 stores)
// ---------------------------------------------------------------------------
__global__ __launch_bounds__(256) void gnn_zero_kernel(float4* __restrict__ p, int n4) {
    int i = blockIdx.x * 256 + threadIdx.x;
    if (i < n4) p[i] = make_float4(0.f, 0.f, 0.f, 0.f);
}

// ---------------------------------------------------------------------------
// Kernel 1: build fused transposed weight WcT[n][k], k<128 -> W1[k][n],
//           k>=128 -> W2[k-128][n].  One-time 128KB, stays hot in L2.
// ---------------------------------------------------------------------------
__global__ __launch_bounds__(256) void gnn_wt_kernel(const float* __restrict__ W1,
                                                     const float* __restrict__ W2,
                                                     float* __restrict__ WcT) {
    int t = blockIdx.x * 256 + threadIdx.x;          // t in [0, 128*256)
    int n = t >> 8;                                  // output row (col of W)
    int k = t & 255;                                 // fused K index
    float w = (k < D_FEAT) ? W1[(size_t)k * D_FEAT + n]
                           : W2[(size_t)(k - D_FEAT) * D_FEAT + n];
    WcT[(size_t)n * K_FUSED + k] = w;
}

// ---------------------------------------------------------------------------
// Kernel 2: COO SpMM scatter: h[row] += val * feat[col]
// 32 threads per edge, each thread handles 4 contiguous floats (float4 gather
// from L2-resident features, 4 global_atomic_add_f32 into L2-resident h).
// ---------------------------------------------------------------------------
__global__ __launch_bounds__(256) void gnn_scatter_kernel(const float* __restrict__ feat,
                                                          const long long* __restrict__ rows,
                                                          const long long* __restrict__ cols,
                                                          const float* __restrict__ vals,
                                                          float* __restrict__ hn,
                                                          int E) {
    int t = blockIdx.x * 256 + threadIdx.x;          // E*32 = 51.2M < 2^31
    int e = t >> 5;
    int q = t & 31;                                  // 4-float chunk id
    if (e >= E) return;
    long long r = rows[e];
    long long c = cols[e];
    float v = vals[e];
    const float4 f = *(const float4*)&feat[(size_t)c * D_FEAT + q * 4];
    float* dst = &hn[(size_t)r * D_FEAT + q * 4];
    atomicAdd(dst + 0, v * f.x);
    atomicAdd(dst + 1, v * f.y);
    atomicAdd(dst + 2, v * f.z);
    atomicAdd(dst + 3, v * f.w);
}

// ---------------------------------------------------------------------------
// Kernel 3: fused dual-GEMM + bias + leaky ReLU via V_WMMA_F32_16X16X4_F32.
// Block = 256 thr = 8 waves. Block owns 16 rows; wave w owns cols [16w,16w+16).
// Xc[16][256] = [X+h | X*h] staged in LDS; WcT read from global (L2-hot).
// Each wave: 64 chained wmma ops over K=256.
// ---------------------------------------------------------------------------
__global__ __launch_bounds__(256) void gnn_gemm_kernel(const float* __restrict__ feat,
                                                       const float* __restrict__ hn,
                                                       const float* __restrict__ WcT,
                                                       const float* __restrict__ b1,
                                                       const float* __restrict__ b2,
                                                       float* __restrict__ out,
                                                       int N) {
    __shared__ float lds[16 * LDS_STRIDE];
    const int tid = threadIdx.x;
    const int r0  = blockIdx.x * 16;

    // Stage X1 = f + h and X2 = f * h as one fused K=256 row in LDS.
    // 16 rows * 32 float4 = 512 chunks over 256 threads.
    for (int i = tid; i < 16 * (D_FEAT / 4); i += 256) {
        int row = i >> 5;
        int c4  = (i & 31) * 4;
        int g   = r0 + row;
        float4 f = make_float4(0.f, 0.f, 0.f, 0.f);
        float4 h = make_float4(0.f, 0.f, 0.f, 0.f);
        if (g < N) {
            f = *(const float4*)&feat[(size_t)g * D_FEAT + c4];
            h = *(const float4*)&hn[(size_t)g * D_FEAT + c4];
        }
        float4 x1 = make_float4(f.x + h.x, f.y + h.y, f.z + h.z, f.w + h.w);
        float4 x2 = make_float4(f.x * h.x, f.y * h.y, f.z * h.z, f.w * h.w);
        *(float4*)&lds[row * LDS_STRIDE + c4]          = x1;   // K = 0..127
        *(float4*)&lds[row * LDS_STRIDE + D_FEAT + c4] = x2;   // K = 128..255
    }
    __syncthreads();

    const int wave = tid >> 5;
    const int lane = tid & 31;
    const int half = lane >> 4;      // 0: K={k,k+1}, 1: K={k+2,k+3}
    const int mn   = lane & 15;      // A: row M,  B: col N
    const int col  = wave * 16;

    const float* aptr = &lds[mn * LDS_STRIDE + 2 * half];
    const float* bptr = &WcT[(size_t)(col + mn) * K_FUSED + 2 * half];

    v8f acc = {};
#pragma unroll 8
    for (int k = 0; k < K_FUSED; k += 4) {
        v2f a = *(const v2f*)(aptr + k);   // A-frag: 16x4 f32 tile
        v2f b = *(const v2f*)(bptr + k);   // B-frag: 4x16 f32 tile (transposed W)
        // D = A*B + C   ->  v_wmma_f32_16x16x4_f32
        acc = __builtin_amdgcn_wmma_f32_16x16x4_f32(
            /*neg_a=*/false, a, /*neg_b=*/false, b,
            /*c_mod=*/(short)0, acc, /*reuse_a=*/false, /*reuse_b=*/false);
    }

    // Epilogue: bias + leaky ReLU. C/D layout: VGPR i -> row i + 8*half, col mn.
    const float bias = b1[col + mn] + b2[col + mn];
#pragma unroll
    for (int i = 0; i < 8; ++i) {
        int g = r0 + i + 8 * half;
        if (g < N) {
            float pre = acc[i] + bias;
            out[(size_t)g * D_FEAT + col + mn] = (pre >= 0.f) ? pre : NEG_SLOPE * pre;
        }
    }
}

// ---------------------------------------------------------------------------
extern "C" void kernel_launch(void* const* d_in, const int* in_sizes, int n_in,
                              void* d_out, int out_size, void* d_ws, size_t ws_size,
                              hipStream_t stream) {
    const float*      feat  = (const float*)d_in[0];
    const long long*  arows = (const long long*)d_in[1];
    const long long*  acols = (const long long*)d_in[2];
    const float*      avals = (const float*)d_in[3];
    const float*      W1    = (const float*)d_in[4];
    const float*      b1    = (const float*)d_in[5];
    const float*      W2    = (const float*)d_in[6];
    const float*      b2    = (const float*)d_in[7];
    float*            out   = (float*)d_out;

    const int N = in_sizes[0] / D_FEAT;   // 100000
    const int E = in_sizes[3];            // 1600000

    // Workspace layout: h_neigh [N*128 f32] | WcT [128*256 f32]
    float* hn  = (float*)d_ws;
    float* WcT = hn + (size_t)N * D_FEAT;

    // 1) zero h accumulator
    int n4 = (N * D_FEAT) / 4;
    gnn_zero_kernel<<<(n4 + 255) / 256, 256, 0, stream>>>((float4*)hn, n4);

    // 2) build fused transposed weights (128x256)
    gnn_wt_kernel<<<(D_FEAT * K_FUSED) / 256, 256, 0, stream>>>(W1, W2, WcT);

    // 3) SpMM scatter: 32 threads per edge
    long long scat_threads = (long long)E * 32;
    int scat_blocks = (int)((scat_threads + 255) / 256);
    gnn_scatter_kernel<<<scat_blocks, 256, 0, stream>>>(feat, arows, acols, avals, hn, E);

    // 4) fused dual-GEMM + bias + leaky ReLU (WMMA)
    int gemm_blocks = (N + 15) / 16;
    gnn_gemm_kernel<<<gemm_blocks, 256, 0, stream>>>(feat, hn, WcT, b1, b2, out, N);
}